// RotatEDecoder_30674656428511
// MI455X (gfx1250) — compile-verified
//
#include <hip/hip_runtime.h>
#include <math.h>

typedef float v2f __attribute__((ext_vector_type(2)));
typedef float v8f __attribute__((ext_vector_type(8)));

#define FD 128          // feature dim (fixed by reference)
#define RSTRIDE 132     // padded LDS row stride in floats (bank-conflict-free, 16B-store aligned)
#define EDGE_WAVES 2    // waves (edge groups) per block

// Generic -> LDS(addrspace 3) offset for inline-asm async loads.
typedef __attribute__((address_space(3))) const void* lds_cptr_t;
__device__ __forceinline__ unsigned lds_off(const void* p) {
  return (unsigned)(size_t)(lds_cptr_t)p;
}

// ---------------------------------------------------------------------------
// Kernel 1: per-node inverse L2 norm. One wave per row of z (32 lanes x float4).
// ---------------------------------------------------------------------------
__global__ __launch_bounds__(256) void rotate_invnorm_kernel(
    const float* __restrict__ z, float* __restrict__ invn, int N) {
  int gw   = (int)((blockIdx.x * blockDim.x + threadIdx.x) >> 5);  // global wave = node
  int lane = threadIdx.x & 31;
  if (gw >= N) return;  // wave-uniform branch
  float4 v = ((const float4*)(z + (size_t)gw * FD))[lane];
  float s = v.x * v.x + v.y * v.y + v.z * v.z + v.w * v.w;
#pragma unroll
  for (int off = 16; off > 0; off >>= 1) s += __shfl_xor(s, off, 32);
  if (lane == 0) invn[gw] = 1.0f / fmaxf(sqrtf(s), 1e-12f);
}

// ---------------------------------------------------------------------------
// Kernel 2: cos table for all (relation, dim) pairs.
// ---------------------------------------------------------------------------
__global__ __launch_bounds__(256) void rotate_cos_kernel(
    const float* __restrict__ phase, float* __restrict__ ct, int RD) {
  int i = blockIdx.x * blockDim.x + threadIdx.x;
  if (i < RD) ct[i] = cosf(phase[i]);
}

// ---------------------------------------------------------------------------
// Kernel 3: 16 edges per wave. Diagonal of (z_src .* cos) x z_dst^T via
// 32 chained V_WMMA_F32_16X16X4_F32, then scale by invn[src]*invn[dst].
// B matrix (z_dst rows) is DMA'd global->LDS with async loads (ASYNCcnt);
// A matrix needs the cos multiply so it stages through VGPRs.
// ---------------------------------------------------------------------------
__global__ __launch_bounds__(EDGE_WAVES * 32) void rotate_edge_kernel(
    const float* __restrict__ z, const int* __restrict__ esrc,
    const int* __restrict__ edst, const int* __restrict__ etype,
    const float* __restrict__ invn, const float* __restrict__ ct,
    float* __restrict__ out, int E, int nGroups) {
  __shared__ float sA[EDGE_WAVES][16 * RSTRIDE];    // A = z_src .* cos   (16 x 128, padded)
  __shared__ float sB[EDGE_WAVES][16 * RSTRIDE];    // Bt = z_dst rows    (16 x 128, padded)
  __shared__ float sDiag[EDGE_WAVES][8 * 32];       // C/D dump for diagonal extraction

  int lane  = threadIdx.x & 31;
  int wid   = threadIdx.x >> 5;
  int group = blockIdx.x * EDGE_WAVES + wid;
  if (group >= nGroups) return;  // wave-uniform; no block barriers used

  float* A    = &sA[wid][0];
  float* Bt   = &sB[wid][0];
  float* diag = &sDiag[wid][0];
  int eg = group * 16;

  // lanes 0..15 own one edge's indices; broadcast with shfl
  int sIdx = 0, dIdx = 0, tIdx = 0;
  if (lane < 16) {
    int e = eg + lane;
    if (e < E) { sIdx = esrc[e]; dIdx = edst[e]; tIdx = etype[e]; }
  }

  // ---- stage: 16 edges, all 32 lanes cooperatively handle one row (16B/lane)
  for (int m = 0; m < 16; ++m) {
    int s = __shfl(sIdx, m, 32);
    int d = __shfl(dIdx, m, 32);
    int t = __shfl(tIdx, m, 32);
    // B side: async DMA straight into LDS, no VGPR round trip
    {
      const float* gsrc = z + (size_t)d * FD + 4 * lane;
      unsigned lb = lds_off(Bt + m * RSTRIDE + 4 * lane);
      asm volatile("global_load_async_to_lds_b128 %0, %1, off"
                   :: "v"(lb), "v"((unsigned long long)(size_t)gsrc)
                   : "memory");
    }
    // A side: needs the cos multiply, stage through VGPRs
    float4 zs = ((const float4*)(z  + (size_t)s * FD))[lane];
    float4 cp = ((const float4*)(ct + (size_t)t * FD))[lane];
    float4 av = make_float4(zs.x * cp.x, zs.y * cp.y, zs.z * cp.z, zs.w * cp.w);
    *(float4*)(A + m * RSTRIDE + 4 * lane) = av;
  }
  asm volatile("s_wait_asynccnt 0" ::: "memory");  // B rows landed in LDS
  __builtin_amdgcn_wave_barrier();                 // same-wave LDS ops are in-order

  // ---- WMMA fragment addressing (A 16x4 f32 layout, B mirrors it transposed):
  // lanes 0-15:  row = lane,    K pair {k0,  k0+1}
  // lanes 16-31: row = lane-16, K pair {k0+2,k0+3}
  int row  = lane & 15;
  int koff = (lane >> 4) << 1;
  const float* aPtr = A  + row * RSTRIDE + koff;
  const float* bPtr = Bt + row * RSTRIDE + koff;

  v8f c = {0.f, 0.f, 0.f, 0.f, 0.f, 0.f, 0.f, 0.f};
#pragma unroll
  for (int j = 0; j < 32; ++j) {  // K = 128 in chunks of 4
    v2f a = *(const v2f*)(aPtr + 4 * j);  // ds_load_b64 (pairs fuse to ds_load_2addr_b64)
    v2f b = *(const v2f*)(bPtr + 4 * j);
    c = __builtin_amdgcn_wmma_f32_16x16x4_f32(
        /*neg_a=*/false, a, /*neg_b=*/false, b,
        /*c_mod=*/(short)0, c, /*reuse_a=*/false, /*reuse_b=*/false);
  }

  // ---- dump accumulator, pull diagonal: D[m][m] lives in VGPR (m&7),
  // lane m (m<8) or lane m+16 (m>=8)
#pragma unroll
  for (int i = 0; i < 8; ++i) diag[i * 32 + lane] = c[i];
  __builtin_amdgcn_wave_barrier();

  if (lane < 16) {
    int e = eg + lane;
    if (e < E) {
      int dl = (lane < 8) ? lane : (lane + 16);
      float v  = diag[(lane & 7) * 32 + dl];
      float sc = invn[sIdx] * invn[dIdx];
      out[e] = v * sc;
    }
  }
}

// ---------------------------------------------------------------------------
// Inputs (setup_inputs order): z[N,128] f32, edge_index[2,E] int, edge_type[E]
// int, phase_rel[R,128] f32.  Output: [E] f32.
// Workspace: invn[N] floats, cos table[R*128] floats (~656 KB).
// ---------------------------------------------------------------------------
extern "C" void kernel_launch(void* const* d_in, const int* in_sizes, int n_in,
                              void* d_out, int out_size, void* d_ws, size_t ws_size,
                              hipStream_t stream) {
  const float* z     = (const float*)d_in[0];
  const int*   eidx  = (const int*)d_in[1];
  const int*   etype = (const int*)d_in[2];
  const float* phase = (const float*)d_in[3];

  int N  = in_sizes[0] / FD;
  int E  = in_sizes[2];
  int RD = in_sizes[3];

  const int* esrc = eidx;
  const int* edst = eidx + E;

  float* invn = (float*)d_ws;
  float* ctab = invn + N;
  float* out  = (float*)d_out;

  // 1) inverse norms: one wave per node, 8 waves per 256-thread block
  rotate_invnorm_kernel<<<(N + 7) / 8, 256, 0, stream>>>(z, invn, N);
  // 2) cos table
  rotate_cos_kernel<<<(RD + 255) / 256, 256, 0, stream>>>(phase, ctab, RD);
  // 3) edge scores: 16 edges per wave, EDGE_WAVES waves per block
  int nGroups = (E + 15) / 16;
  rotate_edge_kernel<<<(nGroups + EDGE_WAVES - 1) / EDGE_WAVES, EDGE_WAVES * 32, 0, stream>>>(
      z, esrc, edst, etype, invn, ctab, out, E, nGroups);
}